// BaseCaps_36670430773720
// MI455X (gfx1250) — compile-verified
//
#include <hip/hip_runtime.h>
#include <hip/hip_bf16.h>

// EM capsule routing, fused E+M passes. B*S tiles, one workgroup per tile.
// lane == output capsule o (O==32 == wave32), wave strides over input caps i.
// CDNA5 paths: TDM tensor_load_to_lds (in_act staging), global async->LDS
// (beta staging), ds_add_f32 cross-wave moment reduction, wave32 shuffles.

#define CAP_I 288
#define CAP_O 32
#define CAP_P 16
#define EMEPS 1e-7f
#define NTHREADS 512
#define NWAVES (NTHREADS / 32)

typedef unsigned int v4u __attribute__((ext_vector_type(4)));
typedef int v8i __attribute__((ext_vector_type(8)));
typedef int v4i __attribute__((ext_vector_type(4)));

__device__ __forceinline__ unsigned lds_byte_addr(const void* p) {
  // numeric value of an AS(3) pointer is the LDS byte offset
  return (unsigned)(unsigned long long)(__attribute__((address_space(3))) const void*)p;
}

__global__ __launch_bounds__(NTHREADS) void emrouting_kernel(
    const float* __restrict__ in_act,   // [BS, I]
    const float* __restrict__ votes,    // [BS, I, O, P]
    const float* __restrict__ beta_a,   // [O]
    const float* __restrict__ beta_v,   // [O]
    const int*   __restrict__ routings_p,
    float* __restrict__ out_act_g,      // [BS, O]
    float* __restrict__ out_means_g)    // [BS, O, P]
{
  const int tid  = threadIdx.x;
  const int lane = tid & 31;            // == output capsule o
  const int wave = tid >> 5;
  const int tile = blockIdx.x;          // b*S + s

  __shared__ float actlds[CAP_I];
  __shared__ float betalds[2 * CAP_O];             // beta_a[32] | beta_v[32]
  __shared__ float acc[CAP_O + 2 * CAP_O * CAP_P]; // S0[32] | S1[p][o] | S2[p][o]
  __shared__ float c_pl[CAP_O * CAP_P];            // 1/(2*var+eps), [p][o]
  __shared__ float q_pl[CAP_O * CAP_P];            // 2*m*c,        [p][o]
  __shared__ float combo[CAP_O];                   // log(a)+prob_main-sum(m^2 c)

  // ---- stage in_act via the Tensor Data Mover (one DMA per workgroup) ----
  if (wave == 0) {
    const unsigned long long ga =
        (unsigned long long)(const void*)(in_act + (size_t)tile * CAP_I);
    const unsigned laddr = lds_byte_addr(&actlds[0]);
    v4u g0;
    g0[0] = 1u;                                   // count=1, gather off
    g0[1] = laddr;                                // lds_addr [63:32]
    g0[2] = (unsigned)(ga & 0xffffffffu);         // global_addr [95:64]
    g0[3] = (unsigned)((ga >> 32) & 0x01ffffffu)  // global_addr [120:96]
            | 0x80000000u;                        // type=2 [127:126]
    v8i g1;
    g1[0] = (int)(2u << 16);                      // data_size = 4 bytes
    g1[1] = (int)((CAP_I & 0xffffu) << 16);       // tensor_dim0 lo16 [63:48]
    g1[2] = (int)(1u << 16);                      // dim0 hi=0 | tensor_dim1=1 lo16
    g1[3] = (int)((unsigned)CAP_I << 16);         // dim1 hi=0 | tile_dim0=288
    g1[4] = 1;                                    // tile_dim1=1, tile_dim2=0
    g1[5] = CAP_I;                                // tensor_dim0_stride lo32
    g1[6] = (int)((CAP_I & 0xffffu) << 16);       // stride0 hi=0 | stride1 lo16
    g1[7] = 0;                                    // stride1 hi
    v4i gz4 = {0, 0, 0, 0};
    v8i gz8 = {0, 0, 0, 0, 0, 0, 0, 0};
    __builtin_amdgcn_tensor_load_to_lds(g0, g1, gz4, gz4, gz8, 0);
    __builtin_amdgcn_s_wait_tensorcnt(0);
  }

  // ---- stage beta_a|beta_v via CDNA5 async global->LDS copies ----
  if (tid < 2 * CAP_O) {
    const float* src = (tid < CAP_O) ? (beta_a + tid) : (beta_v + (tid - CAP_O));
    unsigned long long gaddr = (unsigned long long)(const void*)src;
    unsigned laddr = lds_byte_addr(&betalds[tid]);
    asm volatile("global_load_async_to_lds_b32 %0, %1, off"
                 :: "v"(laddr), "v"(gaddr) : "memory");
    asm volatile("s_wait_asynccnt 0" ::: "memory");
  }
  __syncthreads();

  const int R = routings_p[0];
  float pw = 1.0f;  // 0.95^it

  const float* vbase = votes + (size_t)tile * CAP_I * CAP_O * CAP_P + lane * CAP_P;

  for (int it = 0; it < R; ++it) {
    const bool first = (it == 0);
    const bool last  = (it == R - 1);

    // zero the moment accumulators (stats of the previous pass were consumed
    // before the barrier that ended the previous pass)
    for (int k = tid; k < CAP_O + 2 * CAP_O * CAP_P; k += NTHREADS) acc[k] = 0.0f;

    // hoist E-step coefficients into registers (valid after prev pass barrier)
    float creg[CAP_P], qreg[CAP_P], combo_r = 0.0f;
    if (!first) {
      combo_r = combo[lane];
#pragma unroll
      for (int p = 0; p < CAP_P; ++p) {
        creg[p] = c_pl[p * CAP_O + lane];
        qreg[p] = q_pl[p * CAP_O + lane];
      }
    }
    __syncthreads();

    // ---- fused E (prev stats) + M accumulation over i ----
    float s0 = 0.0f;
    float s1[CAP_P], s2[CAP_P];
#pragma unroll
    for (int p = 0; p < CAP_P; ++p) { s1[p] = 0.0f; s2[p] = 0.0f; }

    for (int i = wave; i < CAP_I; i += NWAVES) {
      const float4* vp = (const float4*)(vbase + (size_t)i * (CAP_O * CAP_P));
      float v[CAP_P];
      *(float4*)&v[0]  = vp[0];
      *(float4*)&v[4]  = vp[1];
      *(float4*)&v[8]  = vp[2];
      *(float4*)&v[12] = vp[3];
      const float ai = actlds[i];  // wave-uniform LDS broadcast

      float w;
      if (first) {
        w = ai * (1.0f / (float)CAP_O);
      } else {
        // zz = combo + sum_p v*q - sum_p v^2*c
        float e = combo_r;
#pragma unroll
        for (int p = 0; p < CAP_P; ++p) {
          e = fmaf(v[p], qreg[p], e);
          e = fmaf(v[p] * v[p], -creg[p], e);
        }
        // softmax over the 32 lanes (= output capsules)
        float mx = e;
#pragma unroll
        for (int d = 16; d >= 1; d >>= 1) mx = fmaxf(mx, __shfl_xor(mx, d));
        float ex = __expf(e - mx);
        float sm = ex;
#pragma unroll
        for (int d = 16; d >= 1; d >>= 1) sm += __shfl_xor(sm, d);
        w = (ex / sm) * ai;
      }

      s0 += w;
#pragma unroll
      for (int p = 0; p < CAP_P; ++p) {
        float t = w * v[p];
        s1[p] += t;
        s2[p] = fmaf(t, v[p], s2[p]);
      }
    }

    // cross-wave reduction of moments via LDS float atomics (ds_add_f32)
    atomicAdd(&acc[lane], s0);
#pragma unroll
    for (int p = 0; p < CAP_P; ++p) {
      atomicAdd(&acc[CAP_O + p * CAP_O + lane], s1[p]);
      atomicAdd(&acc[CAP_O + CAP_O * CAP_P + p * CAP_O + lane], s2[p]);
    }
    __syncthreads();

    // ---- stats: one lane per output capsule (wave 0) ----
    if (tid < CAP_O) {
      const int o = tid;
      const float S0v = acc[o];
      const float inv_rs = 1.0f / (S0v + EMEPS);
      const float lam = 0.01f * (1.0f - pw);
      float logstd_sum = 0.0f, msqc = 0.0f;
#pragma unroll
      for (int p = 0; p < CAP_P; ++p) {
        const float S1v = acc[CAP_O + p * CAP_O + o];
        const float S2v = acc[CAP_O + CAP_O * CAP_P + p * CAP_O + o];
        const float m   = S1v * inv_rs;
        // sum w (v-m)^2 = S2 - m*(2*S1 - m*S0)
        const float varnum = S2v - m * (2.0f * S1v - m * S0v);
        const float var    = fmaxf(varnum, 0.0f) * inv_rs;
        const float sd     = __builtin_sqrtf(var);
        const float ls     = __logf(sd + EMEPS);
        logstd_sum += ls;
        const float c = 1.0f / (2.0f * var + EMEPS);
        const float q = 2.0f * m * c;
        c_pl[p * CAP_O + o] = c;
        q_pl[p * CAP_O + o] = q;
        msqc = fmaf(m * m, c, msqc);
        if (last) out_means_g[((size_t)tile * CAP_O + o) * CAP_P + p] = m;
      }
      const float costsum = (float)CAP_P * betalds[CAP_O + o] + S0v * logstd_sum;
      const float oa = 1.0f / (1.0f + __expf(-(lam * (betalds[o] - costsum))));
      if (last) out_act_g[(size_t)tile * CAP_O + o] = oa;
      combo[o] = __logf(oa + EMEPS) - logstd_sum - msqc;
    }
    __syncthreads();
    pw *= 0.95f;
  }
}

extern "C" void kernel_launch(void* const* d_in, const int* in_sizes, int n_in,
                              void* d_out, int out_size, void* d_ws, size_t ws_size,
                              hipStream_t stream) {
  const float* in_act  = (const float*)d_in[0];
  const float* votes   = (const float*)d_in[1];
  const float* beta_a  = (const float*)d_in[2];
  const float* beta_v  = (const float*)d_in[3];
  const int*   routing = (const int*)d_in[4];

  const int BS = in_sizes[0] / CAP_I;   // B*S tiles (16*36 = 576)
  float* out       = (float*)d_out;
  float* out_act   = out;               // [BS, O]
  float* out_means = out + (size_t)BS * CAP_O;  // [BS, O, P]

  emrouting_kernel<<<BS, NTHREADS, 0, stream>>>(
      in_act, votes, beta_a, beta_v, routing, out_act, out_means);
}